// GCNIIConvolution_29841432772820
// MI455X (gfx1250) — compile-verified
//
#include <hip/hip_runtime.h>
#include <math.h>

typedef __attribute__((ext_vector_type(2))) float v2f;
typedef __attribute__((ext_vector_type(8))) float v8f;

#define FEAT 64

// ---------------------------------------------------------------------------
// Kernel 1: edge-parallel SPMM.  One wave32 per edge; each lane handles two
// consecutive features (float2) -> 64 features per wave.  Scatter with
// global_atomic_add_f32 into AH (workspace), which stays L2-resident (25.6MB
// of 192MB L2).
// ---------------------------------------------------------------------------
__global__ void gcnii_spmm_edges(const int* __restrict__ row,
                                 const int* __restrict__ col,
                                 const float* __restrict__ vals,
                                 const float* __restrict__ H,
                                 float* __restrict__ AH,
                                 int nEdges) {
    int wave = (int)((blockIdx.x * blockDim.x + threadIdx.x) >> 5);
    int lane = threadIdx.x & 31;
    if (wave >= nEdges) return;

    int r = row[wave];
    int c = col[wave];
    float v = vals[wave];

    const float2* hsrc = (const float2*)(H + (size_t)c * FEAT);
    float2 h = hsrc[lane];                    // coalesced 256B per wave

    float* dst = AH + (size_t)r * FEAT + lane * 2;
    atomicAdd(dst + 0, v * h.x);              // global_atomic_add_f32
    atomicAdd(dst + 1, v * h.y);
}

// ---------------------------------------------------------------------------
// Kernel 2: fused support-build + (support @ weight) via V_WMMA_F32_16X16X4_F32
// + epilogue.  One wave per 16x16 output tile; K=64 accumulated in 16 WMMA
// steps.  support = (1-alpha)*AH + alpha*H0 is built on the fly for both the
// A-fragments and the epilogue (AH/H0 re-reads hit L2).
//
// A 16x4 f32 layout (ISA 7.12.2): lanes 0-15 = rows M, v0=K0+0, v1=K0+1;
//                                 lanes 16-31 = rows M, v0=K0+2, v1=K0+3.
// B 4x16 mirrored with N across lanes.  C/D: VGPR r -> (M=r, N=lane) for
// lanes 0-15 and (M=r+8, N=lane-16) for lanes 16-31.
// ---------------------------------------------------------------------------
__global__ void gcnii_gemm_epilogue(const float* __restrict__ AH,
                                    const float* __restrict__ H0,
                                    const float* __restrict__ W,
                                    float* __restrict__ out,
                                    const float* __restrict__ lamdaPtr,
                                    const float* __restrict__ alphaPtr,
                                    const int* __restrict__ lPtr,
                                    int nTiles) {
    int tile = (int)(blockIdx.x * (blockDim.x >> 5) + (threadIdx.x >> 5));
    if (tile >= nTiles) return;

    int m0 = (tile >> 2) * 16;       // row-tile base
    int n0 = (tile & 3) * 16;        // col-tile base (D=64 -> 4 tiles)
    int lane = threadIdx.x & 31;
    int hi   = lane >> 4;            // 0: lanes 0-15, 1: lanes 16-31
    int lm   = lane & 15;

    float alpha = alphaPtr[0];
    float lamda = lamdaPtr[0];
    float beta  = logf(lamda / (float)lPtr[0] + 1.0f);
    float oma   = 1.0f - alpha;
    float omb   = 1.0f - beta;

    const float* arowAH = AH + (size_t)(m0 + lm) * FEAT + hi * 2;
    const float* arowH0 = H0 + (size_t)(m0 + lm) * FEAT + hi * 2;
    const float* wbase  = W + (size_t)(hi * 2) * FEAT + n0 + lm;

    v8f acc = {};
#pragma unroll
    for (int kk = 0; kk < 16; ++kk) {
        int K0 = kk * 4;
        float2 ah2 = *(const float2*)(arowAH + K0);
        float2 h02 = *(const float2*)(arowH0 + K0);
        v2f a;
        a.x = oma * ah2.x + alpha * h02.x;   // support, on the fly
        a.y = oma * ah2.y + alpha * h02.y;
        v2f b;
        b.x = wbase[(size_t)(K0 + 0) * FEAT];
        b.y = wbase[(size_t)(K0 + 1) * FEAT];
        // D = A(16x4) * B(4x16) + C   -> v_wmma_f32_16x16x4_f32
        acc = __builtin_amdgcn_wmma_f32_16x16x4_f32(
            /*neg_a=*/false, a, /*neg_b=*/false, b,
            /*c_mod=*/(short)0, acc, /*reuse_a=*/false, /*reuse_b=*/false);
    }

#pragma unroll
    for (int r = 0; r < 8; ++r) {
        size_t idx = (size_t)(m0 + r + hi * 8) * FEAT + n0 + lm;
        float sup = oma * AH[idx] + alpha * H0[idx];
        out[idx] = omb * sup + beta * acc[r];
    }
}

// ---------------------------------------------------------------------------
// Launch.  Inputs (setup_inputs order):
//   0:row i32[E] 1:col i32[E] 2:vals f32[E] 3:H f32[N*64] 4:H0 f32[N*64]
//   5:weight f32[64*64] 6:lamda f32[1] 7:alpha f32[1] 8:l i32[1]
// Workspace: AH accumulator, N*64 f32 = 25.6 MB.
// ---------------------------------------------------------------------------
extern "C" void kernel_launch(void* const* d_in, const int* in_sizes, int n_in,
                              void* d_out, int out_size, void* d_ws, size_t ws_size,
                              hipStream_t stream) {
    const int*   row   = (const int*)  d_in[0];
    const int*   col   = (const int*)  d_in[1];
    const float* vals  = (const float*)d_in[2];
    const float* H     = (const float*)d_in[3];
    const float* H0    = (const float*)d_in[4];
    const float* W     = (const float*)d_in[5];
    const float* lamda = (const float*)d_in[6];
    const float* alpha = (const float*)d_in[7];
    const int*   l     = (const int*)  d_in[8];

    int nEdges = in_sizes[0];
    int nNodes = in_sizes[3] / FEAT;

    float* AH = (float*)d_ws;

    // Zero the accumulator (capture-safe).
    hipMemsetAsync(AH, 0, (size_t)nNodes * FEAT * sizeof(float), stream);

    // SPMM: one wave per edge, 8 waves (256 threads) per block.
    int spmmBlocks = (nEdges + 7) / 8;
    gcnii_spmm_edges<<<spmmBlocks, 256, 0, stream>>>(row, col, vals, H, AH, nEdges);

    // GEMM + epilogue: one wave per 16x16 tile, 8 waves per block.
    int nTiles = (nNodes / 16) * (FEAT / 16);       // 6250 * 4 = 25000
    int gemmBlocks = (nTiles + 7) / 8;              // 3125
    gcnii_gemm_epilogue<<<gemmBlocks, 256, 0, stream>>>(
        AH, H0, W, (float*)d_out, lamda, alpha, l, nTiles);
}